// HyperbolicAttH_69303592288544
// MI455X (gfx1250) — compile-verified
//
#include <hip/hip_runtime.h>
#include <math.h>
#include <stdint.h>

typedef __attribute__((ext_vector_type(2))) float v2f;
typedef __attribute__((ext_vector_type(8))) float v8f;
typedef __attribute__((ext_vector_type(4))) int   v4i;

#define C_CURV   0.01f
#define SQRT_C   0.1f
#define EPS_F    1e-15f
#define MIN_NORM 1e-10f
#define D_DIM    256

#define BM 64
#define BN 64
#define KC 64
#define NCH (D_DIM / KC)  // 4 K-chunks
#define LDSS 68   // padded LDS row stride (dwords): 68 % 64 == 4 -> conflict-free b64 frag reads

// ---- CDNA5 async Global->LDS path (guarded; falls back to sync staging) ----
#if __has_builtin(__builtin_amdgcn_global_load_async_to_lds_b128) && \
    __has_builtin(__builtin_amdgcn_s_wait_asynccnt)
#define USE_ASYNC_LDS 1
#else
#define USE_ASYNC_LDS 0
#endif

typedef __attribute__((address_space(1))) v4i as1_v4i;  // global int4
typedef __attribute__((address_space(3))) v4i as3_v4i;  // LDS int4

__device__ __forceinline__ as3_v4i* lds_v4i(void* p) {
  // flat LDS pointer carries the LDS byte offset in its low 32 bits
  return (as3_v4i*)(uint32_t)(uintptr_t)p;
}
__device__ __forceinline__ as1_v4i* glb_v4i(const void* p) {
  // integer round-trip performs the addrspace cast and sheds constness
  return (as1_v4i*)(uintptr_t)p;
}

__device__ __forceinline__ float wave_sum(float v) {
#pragma unroll
  for (int off = 16; off >= 1; off >>= 1) v += __shfl_xor(v, off, 32);
  return v;
}

// ---------------------------------------------------------------------------
// Kernel 1: per-query hyperbolic pipeline. One wave32 per query row.
// Emits xq = -query (row-major B x D), x2q[b] = |xq|^2, sb[b] = entity_bias[s].
// ---------------------------------------------------------------------------
__global__ __launch_bounds__(256) void build_query_kernel(
    const float* __restrict__ ent, const float* __restrict__ rel_emb,
    const float* __restrict__ rel_rot, const float* __restrict__ rel_ref,
    const float* __restrict__ attn_w, const float* __restrict__ rel_trans,
    const float* __restrict__ ent_bias, const int* __restrict__ trip,
    float* __restrict__ xq, float* __restrict__ x2q, float* __restrict__ sb,
    int B) {
  const int lane = threadIdx.x & 31;
  const int wave = threadIdx.x >> 5;
  const int b = blockIdx.x * 8 + wave;
  if (b >= B) return;
  const int s = trip[b * 3 + 0];
  const int r = trip[b * 3 + 1];
  const int d0 = lane * 8;

  float sv[8], st[8];
  float acc = 0.f;
#pragma unroll
  for (int j = 0; j < 8; ++j) {
    sv[j] = ent[(size_t)s * D_DIM + d0 + j];
    acc += sv[j] * sv[j];
  }
  // log_map_zero
  float n2 = wave_sum(acc);
  float norm = fmaxf(sqrtf(n2), MIN_NORM);
  float arg = fminf(SQRT_C * norm, 1.0f - 1e-5f);
  float fac = atanhf(arg) / (SQRT_C * norm);
#pragma unroll
  for (int j = 0; j < 8; ++j) st[j] = fac * sv[j];

  // Givens rotation / reflection on (even, odd) pairs
  float rot[8], ref[8];
#pragma unroll
  for (int j = 0; j < 8; j += 2) {
    const int p = (d0 + j) >> 1;
    const float ar = rel_rot[(size_t)r * (D_DIM / 2) + p];
    const float af = rel_ref[(size_t)r * (D_DIM / 2) + p];
    const float x1 = st[j], x2v = st[j + 1];
    const float ca = cosf(ar), sa = sinf(ar);
    rot[j]     = ca * x1 - sa * x2v;
    rot[j + 1] = sa * x1 + ca * x2v;
    const float cb = cosf(af), sbv = sinf(af);
    ref[j]     = cb * x1 + sbv * x2v;
    ref[j + 1] = sbv * x1 - cb * x2v;
  }

  // attention scalar: sigmoid(w . [s_tan, rel_emb])
  float ap = 0.f;
#pragma unroll
  for (int j = 0; j < 8; ++j) {
    ap += attn_w[(size_t)r * (2 * D_DIM) + d0 + j] * st[j];
    ap += attn_w[(size_t)r * (2 * D_DIM) + D_DIM + d0 + j] *
          rel_emb[(size_t)r * D_DIM + d0 + j];
  }
  const float a_r = 1.f / (1.f + expf(-wave_sum(ap)));

  // mix + exp_map_zero
  float mx[8];
  acc = 0.f;
#pragma unroll
  for (int j = 0; j < 8; ++j) {
    mx[j] = a_r * rot[j] + (1.f - a_r) * ref[j];
    acc += mx[j] * mx[j];
  }
  n2 = wave_sum(acc);
  norm = fmaxf(sqrtf(n2), MIN_NORM);
  arg = fminf(SQRT_C * norm, 15.f);
  fac = tanhf(arg) / (SQRT_C * norm);
  float mh[8];
#pragma unroll
  for (int j = 0; j < 8; ++j) mh[j] = fac * mx[j];

  // t_r = exp_map_zero(rel_trans[r])
  float tv[8];
  acc = 0.f;
#pragma unroll
  for (int j = 0; j < 8; ++j) {
    tv[j] = rel_trans[(size_t)r * D_DIM + d0 + j];
    acc += tv[j] * tv[j];
  }
  n2 = wave_sum(acc);
  norm = fmaxf(sqrtf(n2), MIN_NORM);
  arg = fminf(SQRT_C * norm, 15.f);
  fac = tanhf(arg) / (SQRT_C * norm);
#pragma unroll
  for (int j = 0; j < 8; ++j) tv[j] *= fac;

  // mobius_add(mh, tv)
  float axy = 0.f, ax2 = 0.f, ay2 = 0.f;
#pragma unroll
  for (int j = 0; j < 8; ++j) {
    axy += mh[j] * tv[j];
    ax2 += mh[j] * mh[j];
    ay2 += tv[j] * tv[j];
  }
  const float xy = wave_sum(axy);
  const float x2m = wave_sum(ax2);
  const float y2m = wave_sum(ay2);
  const float cx = 1.f + 2.f * C_CURV * xy + C_CURV * y2m;
  const float cy = 1.f - C_CURV * x2m;
  const float den = 1.f + 2.f * C_CURV * xy + C_CURV * C_CURV * x2m * y2m + EPS_F;

  float q2 = 0.f;
#pragma unroll
  for (int j = 0; j < 8; ++j) {
    const float q = (cx * mh[j] + cy * tv[j]) / den;
    const float xv = -q;  // x = -query
    xq[(size_t)b * D_DIM + d0 + j] = xv;
    q2 += xv * xv;
  }
  const float qn = wave_sum(q2);
  if (lane == 0) {
    x2q[b] = qn;
    sb[b] = ent_bias[s];
  }
}

// ---------------------------------------------------------------------------
// Kernel 2: per-entity squared norm. One wave32 per entity row.
// ---------------------------------------------------------------------------
__global__ __launch_bounds__(256) void y2_kernel(const float* __restrict__ ent,
                                                 float* __restrict__ y2, int NE) {
  const int lane = threadIdx.x & 31;
  const int wave = threadIdx.x >> 5;
  const int n = blockIdx.x * 8 + wave;
  if (n >= NE) return;
  float acc = 0.f;
#pragma unroll
  for (int j = 0; j < 8; ++j) {
    const float v = ent[(size_t)n * D_DIM + lane * 8 + j];
    acc += v * v;
  }
  const float s = wave_sum(acc);
  if (lane == 0) y2[n] = s;
}

// ---------------------------------------------------------------------------
// Kernel 3: fused WMMA GEMM (xy = x @ E^T, fp32, v_wmma_f32_16x16x4_f32) +
// Mobius-distance epilogue. Double-buffered LDS; on CDNA5 the staging uses
// GLOBAL_LOAD_ASYNC_TO_LDS_B128 tracked by ASYNCcnt (s_wait_asynccnt), so the
// next K-chunk streams into LDS while the current one feeds the WMMAs.
// ---------------------------------------------------------------------------
__global__ __launch_bounds__(256) void atth_score_kernel(
    const float* __restrict__ xq, const float* __restrict__ ent,
    const float* __restrict__ x2q, const float* __restrict__ sb,
    const float* __restrict__ y2, const float* __restrict__ ent_bias,
    float* __restrict__ out, int NE, int B) {
  __shared__ float sX[2][BM * LDSS];
  __shared__ float sE[2][BN * LDSS];

  const int tid = threadIdx.x;
  const int lane = tid & 31;
  const int wave = tid >> 5;
  const int wm = wave & 3;
  const int wn = wave >> 2;
  const int n0 = blockIdx.x * BN;
  const int m0 = blockIdx.y * BM;

  v8f acc0 = {};
  v8f acc1 = {};

  // fragment addressing (ISA 7.12.2, 32-bit A 16x4 / B 4x16 / C 16x16):
  const int am = wm * 16 + (lane & 15);          // A row in LDS
  const int bn = wn * 32 + (lane & 15);          // B row (entity) in LDS, tile0
  const int chi = (lane >> 4) * 2;               // K sub-offset per half-wave

#if USE_ASYNC_LDS
  // ---- async staging: 8 x b128 per wave per chunk (4 X slots + 4 E slots) --
#define STAGE_ASYNC(BSEL, KCHUNK)                                              \
  {                                                                            \
    _Pragma("unroll") for (int i = 0; i < 4; ++i) {                            \
      const int slot = tid + i * 256;                                          \
      const int row = slot >> 4;                                               \
      const int c4 = slot & 15;                                                \
      if (m0 + row < B)                                                        \
        __builtin_amdgcn_global_load_async_to_lds_b128(                        \
            glb_v4i(&xq[(size_t)(m0 + row) * D_DIM + (KCHUNK) + c4 * 4]),      \
            lds_v4i(&sX[(BSEL)][row * LDSS + c4 * 4]), 0, 0);                  \
      if (n0 + row < NE)                                                       \
        __builtin_amdgcn_global_load_async_to_lds_b128(                        \
            glb_v4i(&ent[(size_t)(n0 + row) * D_DIM + (KCHUNK) + c4 * 4]),     \
            lds_v4i(&sE[(BSEL)][row * LDSS + c4 * 4]), 0, 0);                  \
    }                                                                          \
  }

  STAGE_ASYNC(0, 0);
  STAGE_ASYNC(1, KC);

#pragma unroll
  for (int c = 0; c < NCH; ++c) {
    // chunk c's 8 async ops are the oldest; chunk c+1's 8 may remain in flight
    if (c + 1 < NCH)
      __builtin_amdgcn_s_wait_asynccnt(8);
    else
      __builtin_amdgcn_s_wait_asynccnt(0);
    __syncthreads();

    const float* bx = sX[c & 1];
    const float* be = sE[c & 1];
#pragma unroll
    for (int ks = 0; ks < KC; ks += 4) {
      const int col = ks + chi;
      const v2f a  = *reinterpret_cast<const v2f*>(&bx[am * LDSS + col]);
      const v2f b0 = *reinterpret_cast<const v2f*>(&be[bn * LDSS + col]);
      const v2f b1 = *reinterpret_cast<const v2f*>(&be[(bn + 16) * LDSS + col]);
      acc0 = __builtin_amdgcn_wmma_f32_16x16x4_f32(false, a, false, b0,
                                                   (short)0, acc0, false, false);
      acc1 = __builtin_amdgcn_wmma_f32_16x16x4_f32(false, a, false, b1,
                                                   (short)0, acc1, false, false);
    }
    __syncthreads();  // all waves done reading buf[c&1] before refill

    if (c + 2 < NCH) STAGE_ASYNC(c & 1, (c + 2) * KC);
  }
#undef STAGE_ASYNC

#else  // ---------------- synchronous fallback staging ----------------------
  for (int kc = 0; kc < D_DIM; kc += KC) {
    if (kc + KC < D_DIM) {
      const int row = (tid & 127) >> 1;
      const int half = (tid & 1) * 32;
      if (tid < 128) {
        if (m0 + row < B)
          __builtin_prefetch(&xq[(size_t)(m0 + row) * D_DIM + kc + KC + half], 0, 0);
      } else {
        if (n0 + row < NE)
          __builtin_prefetch(&ent[(size_t)(n0 + row) * D_DIM + kc + KC + half], 0, 0);
      }
    }
#pragma unroll
    for (int i = 0; i < 4; ++i) {
      const int slot = tid + i * 256;
      const int row = slot >> 4;
      const int c4 = slot & 15;
      float4 vx = {0.f, 0.f, 0.f, 0.f};
      if (m0 + row < B)
        vx = *reinterpret_cast<const float4*>(
            &xq[(size_t)(m0 + row) * D_DIM + kc + c4 * 4]);
      *reinterpret_cast<float4*>(&sX[0][row * LDSS + c4 * 4]) = vx;
      float4 ve = {0.f, 0.f, 0.f, 0.f};
      if (n0 + row < NE)
        ve = *reinterpret_cast<const float4*>(
            &ent[(size_t)(n0 + row) * D_DIM + kc + c4 * 4]);
      *reinterpret_cast<float4*>(&sE[0][row * LDSS + c4 * 4]) = ve;
    }
    __syncthreads();
#pragma unroll
    for (int ks = 0; ks < KC; ks += 4) {
      const int col = ks + chi;
      const v2f a  = *reinterpret_cast<const v2f*>(&sX[0][am * LDSS + col]);
      const v2f b0 = *reinterpret_cast<const v2f*>(&sE[0][bn * LDSS + col]);
      const v2f b1 = *reinterpret_cast<const v2f*>(&sE[0][(bn + 16) * LDSS + col]);
      acc0 = __builtin_amdgcn_wmma_f32_16x16x4_f32(false, a, false, b0,
                                                   (short)0, acc0, false, false);
      acc1 = __builtin_amdgcn_wmma_f32_16x16x4_f32(false, a, false, b1,
                                                   (short)0, acc1, false, false);
    }
    __syncthreads();
  }
#endif

  // Epilogue: element (vgpr i, lane L): M = i + 8*(L>=16), N = L%16
  const int mbase = m0 + wm * 16 + ((lane >> 4) << 3);
  const int nn0 = n0 + wn * 32 + (lane & 15);
  const int nn1 = nn0 + 16;
  const float y2a = (nn0 < NE) ? y2[nn0] : 0.f;
  const float eba = (nn0 < NE) ? ent_bias[nn0] : 0.f;
  const float y2b = (nn1 < NE) ? y2[nn1] : 0.f;
  const float ebb = (nn1 < NE) ? ent_bias[nn1] : 0.f;

#pragma unroll
  for (int i = 0; i < 8; ++i) {
    const int m = mbase + i;
    if (m >= B) continue;
    const float x2v = x2q[m];
    const float sbv = sb[m];
    const float Bf = 1.f - C_CURV * x2v;
    if (nn0 < NE) {
      const float xyv = acc0[i];
      const float A = 1.f + 2.f * C_CURV * xyv + C_CURV * y2a;
      const float den = 1.f + 2.f * C_CURV * xyv + C_CURV * C_CURV * x2v * y2a + EPS_F;
      const float dist = (A * A * x2v + 2.f * A * Bf * xyv + Bf * Bf * y2a) / (den * den);
      out[(size_t)m * NE + nn0] = -dist + eba + sbv;
    }
    if (nn1 < NE) {
      const float xyv = acc1[i];
      const float A = 1.f + 2.f * C_CURV * xyv + C_CURV * y2b;
      const float den = 1.f + 2.f * C_CURV * xyv + C_CURV * C_CURV * x2v * y2b + EPS_F;
      const float dist = (A * A * x2v + 2.f * A * Bf * xyv + Bf * Bf * y2b) / (den * den);
      out[(size_t)m * NE + nn1] = -dist + ebb + sbv;
    }
  }
}

// ---------------------------------------------------------------------------
extern "C" void kernel_launch(void* const* d_in, const int* in_sizes, int n_in,
                              void* d_out, int out_size, void* d_ws, size_t ws_size,
                              hipStream_t stream) {
  const float* ent       = (const float*)d_in[0];
  const float* rel_emb   = (const float*)d_in[1];
  const float* rel_rot   = (const float*)d_in[2];
  const float* rel_ref   = (const float*)d_in[3];
  const float* attn_w    = (const float*)d_in[4];
  const float* rel_trans = (const float*)d_in[5];
  const float* ent_bias  = (const float*)d_in[6];
  const int*   trip      = (const int*)d_in[7];

  const int NE = in_sizes[0] / D_DIM;
  const int B  = in_sizes[7] / 3;

  float* ws  = (float*)d_ws;
  float* xq  = ws;                          // B * D
  float* x2q = xq + (size_t)B * D_DIM;      // B
  float* sbv = x2q + B;                     // B
  float* y2  = sbv + B;                     // NE
  float* out = (float*)d_out;

  build_query_kernel<<<(B + 7) / 8, 256, 0, stream>>>(
      ent, rel_emb, rel_rot, rel_ref, attn_w, rel_trans, ent_bias, trip,
      xq, x2q, sbv, B);
  y2_kernel<<<(NE + 7) / 8, 256, 0, stream>>>(ent, y2, NE);

  dim3 grid((NE + BN - 1) / BN, (B + BM - 1) / BM);
  atth_score_kernel<<<grid, 256, 0, stream>>>(xq, ent, x2q, sbv, y2, ent_bias,
                                              out, NE, B);
}